// GCNComm_19902878449952
// MI455X (gfx1250) — compile-verified
//
#include <hip/hip_runtime.h>
#include <hip/hip_bf16.h>

#define GB 4
#define GN 8192
#define GD 64
#define MAXDEG 128
#define BN (GB * GN)

typedef float v2f __attribute__((ext_vector_type(2)));
typedef float v8f __attribute__((ext_vector_type(8)));

// ---------------------------------------------------------------------------
// Kernel 1: stream the 1 GiB dense adjacency ONCE (non-temporal), extract a
// deterministic, column-ordered edge list per row, and compute deg^{-1/2}.
// One wave (32 lanes) per row; ballot+popcount gives ordered slot assignment.
// ---------------------------------------------------------------------------
__global__ __launch_bounds__(256) void gcn_edges_kernel(
    const float* __restrict__ adj, int* __restrict__ cols,
    int* __restrict__ degA, float* __restrict__ dinv) {
  int wid  = (blockIdx.x * blockDim.x + threadIdx.x) >> 5;  // global row id
  int lane = threadIdx.x & 31;
  if (wid >= BN) return;
  int i = wid & (GN - 1);  // local row within batch
  const float* ar = adj + (size_t)wid * GN;
  int* cp = cols + (size_t)wid * MAXDEG;
  unsigned lmask = (1u << lane) - 1u;
  int base = 0;  // uniform across wave
  for (int c0 = 0; c0 < GN; c0 += 128) {
#pragma unroll
    for (int e = 0; e < 4; ++e) {
      int col = c0 + e * 32 + lane;
      float v = __builtin_nontemporal_load(&ar[col]);
      bool nz = (v != 0.0f) || (col == i);  // forced self-loop on diagonal
      unsigned m = __builtin_amdgcn_ballot_w32(nz);
      if (nz) {
        int slot = base + __builtin_popcount(m & lmask);
        if (slot < MAXDEG) cp[slot] = col;
      }
      base += __builtin_popcount(m);
    }
  }
  if (lane == 0) {
    int d = base;
    degA[wid] = (d > MAXDEG) ? MAXDEG : d;
    dinv[wid] = rsqrtf((float)d);  // deg >= 1 always (self loop)
  }
}

// ---------------------------------------------------------------------------
// Kernel 2: H = X @ W  (X: [BN,64], W: [64,64]) via V_WMMA_F32_16X16X4_F32.
// One wave computes a 16x64 output strip: 4 f32 accumulators, 16 K-steps,
// 64 WMMA per wave. W staged in LDS (stride 65 to dodge bank conflicts).
// ---------------------------------------------------------------------------
__global__ __launch_bounds__(256) void gcn_gemm64_kernel(
    const float* __restrict__ X, const float* __restrict__ W,
    float* __restrict__ H) {
  __shared__ float sW[64 * 65];
  for (int t = threadIdx.x; t < 64 * 64; t += 256)
    sW[(t >> 6) * 65 + (t & 63)] = W[t];
  __syncthreads();

  int wave = threadIdx.x >> 5;
  int lane = threadIdx.x & 31;
  int rowbase = (blockIdx.x * 8 + wave) * 16;
  if (rowbase >= BN) return;

  const float* Xr = X + (size_t)rowbase * 64;
  int mrow  = lane & 15;            // M (for A) / N (for B,D) index
  int khalf = (lane >> 4) << 1;     // 0 for lanes 0-15, 2 for lanes 16-31

  v8f acc0 = {}, acc1 = {}, acc2 = {}, acc3 = {};
  for (int k0 = 0; k0 < 64; k0 += 4) {
    // A fragment 16x4: a.x = X[rowbase+m][k0+khalf], a.y = [k0+khalf+1]
    const float* ap = Xr + (size_t)mrow * 64 + k0 + khalf;
    v2f a; a.x = ap[0]; a.y = ap[1];
    // B fragments 4x16 from LDS: b.x = W[k0+khalf][col], b.y = W[k0+khalf+1][col]
    const float* b0p = &sW[(k0 + khalf) * 65];
    v2f b0, b1, b2, b3;
    b0.x = b0p[mrow];      b0.y = b0p[65 + mrow];
    b1.x = b0p[16 + mrow]; b1.y = b0p[65 + 16 + mrow];
    b2.x = b0p[32 + mrow]; b2.y = b0p[65 + 32 + mrow];
    b3.x = b0p[48 + mrow]; b3.y = b0p[65 + 48 + mrow];
    acc0 = __builtin_amdgcn_wmma_f32_16x16x4_f32(false, a, false, b0, (short)0, acc0, false, false);
    acc1 = __builtin_amdgcn_wmma_f32_16x16x4_f32(false, a, false, b1, (short)0, acc1, false, false);
    acc2 = __builtin_amdgcn_wmma_f32_16x16x4_f32(false, a, false, b2, (short)0, acc2, false, false);
    acc3 = __builtin_amdgcn_wmma_f32_16x16x4_f32(false, a, false, b3, (short)0, acc3, false, false);
  }

  // D layout: VGPR e -> (M=e, N=lane) lanes 0-15 ; (M=e+8, N=lane-16) lanes 16-31
  float* Hr = H + (size_t)rowbase * 64;
  int roff = (lane >> 4) * 8;
#pragma unroll
  for (int e = 0; e < 8; ++e) {
    int r = e + roff;
    Hr[(size_t)r * 64 +      mrow] = acc0[e];
    Hr[(size_t)r * 64 + 16 + mrow] = acc1[e];
    Hr[(size_t)r * 64 + 32 + mrow] = acc2[e];
    Hr[(size_t)r * 64 + 48 + mrow] = acc3[e];
  }
}

// ---------------------------------------------------------------------------
// Kernel 3: sparse aggregation  out[i] = sum_j dinv_i*dinv_j * H[j]  (+bias,
// optional ELU). One wave per row, each lane owns 2 of the 64 features.
// H (8 MB) is L2-resident on MI455X, so gathers are cheap.
// ---------------------------------------------------------------------------
__global__ __launch_bounds__(256) void gcn_agg_kernel(
    const float* __restrict__ H, const int* __restrict__ cols,
    const int* __restrict__ degA, const float* __restrict__ dinv,
    const float* __restrict__ bias, float* __restrict__ out, int do_elu) {
  int wid  = (blockIdx.x * blockDim.x + threadIdx.x) >> 5;  // global row
  int lane = threadIdx.x & 31;
  if (wid >= BN) return;
  int b = wid >> 13;  // wid / GN
  const int* cp = cols + (size_t)wid * MAXDEG;
  const float* dv = dinv + (size_t)b * GN;
  const float2* Hb = (const float2*)(H + (size_t)b * GN * 64);

  float wi  = dinv[wid];
  int   deg = degA[wid];
  float2 acc = make_float2(0.f, 0.f);

  for (int e0 = 0; e0 < deg; e0 += 32) {
    int n = deg - e0; if (n > 32) n = 32;
    int jv = 0; float wv = 0.f;
    if (lane < n) { jv = cp[e0 + lane]; wv = dv[jv]; }
    for (int t = 0; t < n; ++t) {
      int   jt = __shfl(jv, t, 32);
      float w  = wi * __shfl(wv, t, 32);
      float2 h = Hb[(size_t)jt * 32 + lane];
      acc.x += w * h.x;
      acc.y += w * h.y;
    }
  }
  float rx = acc.x + bias[lane * 2];
  float ry = acc.y + bias[lane * 2 + 1];
  if (do_elu) {
    rx = rx > 0.f ? rx : expm1f(rx);
    ry = ry > 0.f ? ry : expm1f(ry);
  }
  ((float2*)out)[(size_t)wid * 32 + lane] = make_float2(rx, ry);
}

// ---------------------------------------------------------------------------
extern "C" void kernel_launch(void* const* d_in, const int* in_sizes, int n_in,
                              void* d_out, int out_size, void* d_ws, size_t ws_size,
                              hipStream_t stream) {
  const float* x   = (const float*)d_in[0];
  const float* adj = (const float*)d_in[1];
  const float* W1  = (const float*)d_in[2];
  const float* b1  = (const float*)d_in[3];
  const float* W2  = (const float*)d_in[4];
  const float* b2  = (const float*)d_in[5];
  const float* W3  = (const float*)d_in[6];
  const float* b3  = (const float*)d_in[7];
  float* outp = (float*)d_out;

  char* ws = (char*)d_ws;
  int*   cols = (int*)ws;                 ws += (size_t)BN * MAXDEG * sizeof(int);  // 16 MB
  int*   degA = (int*)ws;                 ws += (size_t)BN * sizeof(int);           // 128 KB
  float* dinv = (float*)ws;               ws += (size_t)BN * sizeof(float);         // 128 KB
  float* Hbuf = (float*)ws;               ws += (size_t)BN * GD * sizeof(float);    // 8 MB
  float* Gbuf = (float*)ws;                                                         // 8 MB

  dim3 blk(256);
  dim3 grid_rows(BN / 8);    // 8 waves (rows) per 256-thread block
  dim3 grid_gemm(BN / 128);  // 8 strips of 16 rows per block

  // 1) one streaming pass over the 1 GiB adjacency -> edge list + dinv
  gcn_edges_kernel<<<grid_rows, blk, 0, stream>>>(adj, cols, degA, dinv);

  // 2) layer 1: H = x@W1 (WMMA), then aggregate + bias + ELU
  gcn_gemm64_kernel<<<grid_gemm, blk, 0, stream>>>(x, W1, Hbuf);
  gcn_agg_kernel<<<grid_rows, blk, 0, stream>>>(Hbuf, cols, degA, dinv, b1, Gbuf, 1);

  // 3) layer 2
  gcn_gemm64_kernel<<<grid_gemm, blk, 0, stream>>>(Gbuf, W2, Hbuf);
  gcn_agg_kernel<<<grid_rows, blk, 0, stream>>>(Hbuf, cols, degA, dinv, b2, Gbuf, 1);

  // 4) layer 3 (no activation) -> d_out
  gcn_gemm64_kernel<<<grid_gemm, blk, 0, stream>>>(Gbuf, W3, Hbuf);
  gcn_agg_kernel<<<grid_rows, blk, 0, stream>>>(Hbuf, cols, degA, dinv, b3, outp, 0);
}